// SAGEConv_71622874628186
// MI455X (gfx1250) — compile-verified
//
#include <hip/hip_runtime.h>
#include <hip/hip_bf16.h>

// ---------------------------------------------------------------------------
// SAGEConv (with self-loops, degree-norm, segment-mean) for MI455X / gfx1250.
//   x:        [N, 64]  f32
//   senders:  [E]      i32
//   receivers:[E]      i32
//   W:        [128,64] f32
//   b:        [64]     f32
//   out:      [N, 64]  f32
// ---------------------------------------------------------------------------

typedef float v2f __attribute__((ext_vector_type(2)));
typedef float v8f __attribute__((ext_vector_type(8)));

#define D_FEAT 64

// --- 1) init degrees to 1 (self-loop) --------------------------------------
__global__ void sage_init_deg(int* __restrict__ deg_s, int* __restrict__ deg_r, int n) {
    int i = blockIdx.x * blockDim.x + threadIdx.x;
    if (i < n) { deg_s[i] = 1; deg_r[i] = 1; }
}

// --- 2) count degrees over edges -------------------------------------------
__global__ void sage_count_deg(const int* __restrict__ senders,
                               const int* __restrict__ receivers,
                               int* __restrict__ deg_s, int* __restrict__ deg_r,
                               int n_edges) {
    int e = blockIdx.x * blockDim.x + threadIdx.x;
    if (e < n_edges) {
        atomicAdd(&deg_s[senders[e]], 1);
        atomicAdd(&deg_r[receivers[e]], 1);
    }
}

// --- 3) summed[i] = x[i] * rsqrt(deg_s[i])   (self-loop contribution) ------
// one float4 per thread: n*16 work items
__global__ void sage_init_sum(const float* __restrict__ x,
                              const int* __restrict__ deg_s,
                              float* __restrict__ summed, int n) {
    int idx = blockIdx.x * blockDim.x + threadIdx.x;
    if (idx >= n * (D_FEAT / 4)) return;
    int i  = idx >> 4;           // node
    int c4 = (idx & 15) << 2;    // feature offset
    float scale = rsqrtf(fmaxf((float)deg_s[i], 1.0f));
    const float4 v = *(const float4*)(x + (size_t)i * D_FEAT + c4);
    float4 o; o.x = v.x * scale; o.y = v.y * scale; o.z = v.z * scale; o.w = v.w * scale;
    *(float4*)(summed + (size_t)i * D_FEAT + c4) = o;
}

// --- 4) scatter: summed[r] += x[s] * rsqrt(deg_s[s]) ------------------------
// 16 threads per edge, one float4 (coalesced 256B row read) + 4 f32 atomics
__global__ __launch_bounds__(256) void sage_scatter(
    const float* __restrict__ x, const int* __restrict__ senders,
    const int* __restrict__ receivers, const int* __restrict__ deg_s,
    float* __restrict__ summed, int n_edges) {
    int idx = blockIdx.x * 256 + threadIdx.x;
    int e = idx >> 4;
    if (e >= n_edges) return;
    int c4 = (idx & 15) << 2;
    int s = senders[e];
    int r = receivers[e];
    float scale = rsqrtf(fmaxf((float)deg_s[s], 1.0f));
    const float4 v = *(const float4*)(x + (size_t)s * D_FEAT + c4);
    float* dst = summed + (size_t)r * D_FEAT + c4;
    atomicAdd(dst + 0, v.x * scale);
    atomicAdd(dst + 1, v.y * scale);
    atomicAdd(dst + 2, v.z * scale);
    atomicAdd(dst + 3, v.w * scale);
}

// --- 5) x_upd = summed / deg_r * rsqrt(deg_r)  (in place) -------------------
__global__ void sage_finalize(float* __restrict__ summed,
                              const int* __restrict__ deg_r, int n) {
    int idx = blockIdx.x * blockDim.x + threadIdx.x;
    if (idx >= n * (D_FEAT / 4)) return;
    int i  = idx >> 4;
    int c4 = (idx & 15) << 2;
    float d = fmaxf((float)deg_r[i], 1.0f);
    float scale = rsqrtf(d) / d;            // d^{-1.5}
    float4 v = *(const float4*)(summed + (size_t)i * D_FEAT + c4);
    v.x *= scale; v.y *= scale; v.z *= scale; v.w *= scale;
    *(float4*)(summed + (size_t)i * D_FEAT + c4) = v;
}

// --- 6) out = [x | x_upd] @ W + b   via V_WMMA_F32_16X16X4_F32 -------------
// One wave computes one 16x16 output tile; K = 128 in 32 steps of 4.
// A-frag (16x4 f32, 2 VGPR): lane L<16 holds A[L][k0..k0+1], lane L+16 holds
//                            A[L][k0+2..k0+3]  (ISA 32-bit A layout)
// B-frag (4x16 f32, 2 VGPR): lane L<16: V0=B[k0][L],  V1=B[k0+1][L]
//                            lane L+16: V0=B[k0+2][L],V1=B[k0+3][L]
// C/D (16x16 f32, 8 VGPR):   VGPR v, lane L<16 -> (row0+v, col0+L),
//                            lane L+16 -> (row0+v+8, col0+L-16+... = col0+L&15)
__global__ __launch_bounds__(256) void sage_gemm_wmma(
    const float* __restrict__ x, const float* __restrict__ xupd,
    const float* __restrict__ W, const float* __restrict__ bvec,
    float* __restrict__ out, int n) {
    __shared__ float Wl[2 * D_FEAT * D_FEAT];   // 128x64 f32 = 32 KB
    for (int i = threadIdx.x; i < 2 * D_FEAT * D_FEAT; i += 256)
        Wl[i] = W[i];
    __syncthreads();

    const int lane  = threadIdx.x & 31;
    const int wave  = threadIdx.x >> 5;
    const int tile  = blockIdx.x * 8 + wave;    // 8 waves per block
    const int tm    = tile >> 2;                // 4 col-tiles (N=64)
    const int tn    = tile & 3;
    const int row0  = tm * 16;
    const int col0  = tn * 16;
    if (row0 >= n) return;

    const int mrow  = lane & 15;                // M (A) / N (B,C) within tile
    const int khalf = lane >> 4;                // 0: k0..k1, 1: k2..k3

    const float* arow = x    + (size_t)(row0 + mrow) * D_FEAT;
    const float* urow = xupd + (size_t)(row0 + mrow) * D_FEAT;
    const float  bias = bvec[col0 + mrow];

    v8f c = {};
    #pragma unroll
    for (int kk = 0; kk < 32; ++kk) {
        const int karr  = kk * 4;
        const int kbase = karr + 2 * khalf;     // per-lane K start
        v2f a;
        if (karr < D_FEAT) {
            a = *(const v2f*)(arow + kbase);
        } else {
            a = *(const v2f*)(urow + (kbase - D_FEAT));
        }
        v2f bm;
        bm.x = Wl[(kbase + 0) * D_FEAT + col0 + mrow];
        bm.y = Wl[(kbase + 1) * D_FEAT + col0 + mrow];
        // (neg_a, A, neg_b, B, c_mod, C, reuse_a, reuse_b)
        c = __builtin_amdgcn_wmma_f32_16x16x4_f32(
                false, a, false, bm, (short)0, c, false, false);
    }

    float* orow = out + (size_t)(row0 + 8 * khalf) * D_FEAT + col0 + mrow;
    #pragma unroll
    for (int v = 0; v < 8; ++v)
        orow[(size_t)v * D_FEAT] = c[v] + bias;
}

// ---------------------------------------------------------------------------
extern "C" void kernel_launch(void* const* d_in, const int* in_sizes, int n_in,
                              void* d_out, int out_size, void* d_ws, size_t ws_size,
                              hipStream_t stream) {
    const float* x         = (const float*)d_in[0];
    const int*   senders   = (const int*)d_in[1];
    const int*   receivers = (const int*)d_in[2];
    const float* W         = (const float*)d_in[3];
    const float* bvec      = (const float*)d_in[4];
    float*       out       = (float*)d_out;

    const int n       = in_sizes[0] / D_FEAT;   // 100000
    const int n_edges = in_sizes[1];            // 1600000

    // workspace layout (256B aligned): deg_s[n], deg_r[n], summed[n*64]
    char* ws = (char*)d_ws;
    size_t degBytes = (((size_t)n * sizeof(int)) + 255) & ~(size_t)255;
    int*   deg_s  = (int*)ws;
    int*   deg_r  = (int*)(ws + degBytes);
    float* summed = (float*)(ws + 2 * degBytes);

    const int T = 256;
    // 1) degrees start at 1 (self-loop)
    sage_init_deg<<<(n + T - 1) / T, T, 0, stream>>>(deg_s, deg_r, n);
    // 2) accumulate edge degrees
    sage_count_deg<<<(n_edges + T - 1) / T, T, 0, stream>>>(senders, receivers,
                                                            deg_s, deg_r, n_edges);
    // 3) self-loop contribution
    int elems4 = n * (D_FEAT / 4);
    sage_init_sum<<<(elems4 + T - 1) / T, T, 0, stream>>>(x, deg_s, summed, n);
    // 4) edge scatter (16 threads / edge)
    long long scatter_items = (long long)n_edges * 16;
    sage_scatter<<<(int)((scatter_items + T - 1) / T), T, 0, stream>>>(
        x, senders, receivers, deg_s, summed, n_edges);
    // 5) mean + receiver-degree norm (in place)
    sage_finalize<<<(elems4 + T - 1) / T, T, 0, stream>>>(summed, deg_r, n);
    // 6) fused concat-GEMM + bias on the WMMA path
    int tiles = (n / 16) * 4;                   // 6250 * 4 = 25000 (exact)
    sage_gemm_wmma<<<tiles / 8, 256, 0, stream>>>(x, summed, W, bvec, out, n);
}